// SigMMDLoss_24507083391568
// MI455X (gfx1250) — compile-verified
//
#include <hip/hip_runtime.h>
#include <hip/hip_bf16.h>

// SigMMD loss, simplified to closed form:
// per-row reductions S, Q2, Q3 plus endpoints p0,pN give the 5 non-constant
// signature features; time-only features cancel between real and gen.
// Memory-bound: 128 MB single pass -> ~5.5 us floor at 23.3 TB/s.
// Data path: async global->LDS copies (ASYNCcnt) + WMMA f32 reductions.

typedef __attribute__((ext_vector_type(2))) float v2f;
typedef __attribute__((ext_vector_type(4))) float v4f;
typedef __attribute__((ext_vector_type(8))) float v8f;

#define T_DIM 4096
#define B_DIM 4096
#define NTHREADS 256

// Sum vals[0..255] using V_WMMA_F32_16X16X4_F32 with an all-ones B matrix.
// D[m][n] = sum_k A[m][k] + C[m][n]; 4 chained WMMAs accumulate 256 values
// into 16 row-sums (replicated across columns), finished with one xor-16
// shuffle. Must be executed by one full wave (lanes 0..31 all active).
// fp32 A/B -> no precision loss vs a scalar tree reduction.
__device__ __forceinline__ float wave_wmma_sum256(const float* vals, int lane) {
  const int m  = lane & 15;          // A-matrix row for this lane
  const int kb = (lane >> 4) << 1;   // lanes 0-15 hold K=0,1 ; lanes 16-31 hold K=2,3
  v8f acc = {};
  v2f ones;
  ones.x = 1.0f;
  ones.y = 1.0f;
#pragma unroll
  for (int c = 0; c < 4; ++c) {
    v2f a;
    a.x = vals[c * 64 + m * 4 + kb];
    a.y = vals[c * 64 + m * 4 + kb + 1];
    // (neg_a, A, neg_b, B, c_mod, C, reuse_a, reuse_b)
    acc = __builtin_amdgcn_wmma_f32_16x16x4_f32(false, a, false, ones,
                                                (short)0, acc, false, false);
  }
  // lane n<16: VGPR r holds D[r][n] (rows 0-7); lane n>=16: rows 8-15.
  float s = acc[0] + acc[1] + acc[2] + acc[3] + acc[4] + acc[5] + acc[6] + acc[7];
  s += __shfl_xor(s, 16, 32);  // combine rows 0-7 with rows 8-15
  return s;
}

// One block per row (2*B rows total). Stage the 16KB row in LDS via
// GLOBAL_LOAD_ASYNC_TO_LDS_B128 (no VGPR round-trip; ASYNCcnt-tracked),
// compute with conflict-free stride-256 LDS reads, reduce S/Q2/Q3 across
// the block with WMMA, emit 5 features per row.
__global__ void __launch_bounds__(NTHREADS)
sig_row_kernel(const float* __restrict__ real_p, const float* __restrict__ gen_p,
               float* __restrict__ partials) {
  __shared__ v4f row4[T_DIM / 4];
  __shared__ float red[3][NTHREADS];
  float* row = (float*)row4;

  const int t   = threadIdx.x;
  const int bid = blockIdx.x;
  const int arr = bid >> 12;       // 0 = real, 1 = gen
  const int r   = bid & (B_DIM - 1);

  const v4f* src4 =
      (const v4f*)((arr ? gen_p : real_p) + (size_t)r * T_DIM);

#pragma unroll
  for (int i = 0; i < 4; ++i) {
    const int idx = t + NTHREADS * i;  // 16B chunk index, coalesced per wave
    // LDS offset: generic pointers to __shared__ carry the LDS offset in the
    // low 32 bits (flat LDS aperture), which is what VDST expects.
    unsigned int lds_addr = (unsigned int)(uintptr_t)(&row4[idx]);
    unsigned long long gaddr = (unsigned long long)(uintptr_t)(src4 + idx);
    asm volatile("global_load_async_to_lds_b128 %0, %1, off"
                 :
                 : "v"(lds_addr), "v"(gaddr)
                 : "memory");
  }
  // Wait for this wave's async copies, then sync the block.
  asm volatile("s_wait_asynccnt 0" ::: "memory");
  __syncthreads();

  // Per-thread partials over k = t + 256*j (bank-conflict-free, covers 0..4094)
  float S = 0.0f, q2 = 0.0f, q3 = 0.0f;
#pragma unroll
  for (int j = 0; j < 16; ++j) {
    const int k = t + NTHREADS * j;
    if (k < T_DIM - 1) {
      const float a  = row[k];
      const float b  = row[k + 1];
      const float dp = b - a;
      S  += a;
      q2 += dp * dp;
      q3 += dp * dp * dp;
    }
  }
  red[0][t] = S;
  red[1][t] = q2;
  red[2][t] = q3;
  __syncthreads();

  if (t < 32) {  // wave 0, EXEC all-ones -> WMMA legal
    const float Stot = wave_wmma_sum256(red[0], t);
    const float Q2   = wave_wmma_sum256(red[1], t);
    const float Q3   = wave_wmma_sum256(red[2], t);
    if (t == 0) {
      const float p0 = row[0];
      const float pN = row[T_DIM - 1];
      const float dt = 1.0f / (float)(T_DIM - 1);
      const float f1 = pN - p0;                        // level1[1]
      const float f2 = pN - Stot * dt;                 // level2: dt*sum (k+1)dp
      const float f3 = 0.5f * (pN * pN - p0 * p0)      // level2: sum cum_p*dp
                     + 0.5f * Q2 - p0 * (pN - p0);
      float* dst = partials + (size_t)arr * 5 * B_DIM + r;
      dst[0 * B_DIM] = f1;
      dst[1 * B_DIM] = f2;
      dst[2 * B_DIM] = f3;
      dst[3 * B_DIM] = Q2;                             // level3[1]
      dst[4 * B_DIM] = Q3;                             // level4[1]
    }
  }
}

// Single block: reduce each of the 10 (array,feature) columns over 4096 rows
// (same WMMA reduction), then combine into the scalar loss.
__global__ void __launch_bounds__(NTHREADS)
sig_finalize_kernel(const float* __restrict__ partials, float* __restrict__ out) {
  __shared__ float red[NTHREADS];
  __shared__ float results[10];
  const int t = threadIdx.x;

  for (int c = 0; c < 10; ++c) {
    const float* base = partials + (size_t)c * B_DIM;
    float s = 0.0f;
#pragma unroll
    for (int j = 0; j < 16; ++j) s += base[t + NTHREADS * j];
    red[t] = s;
    __syncthreads();
    if (t < 32) {
      const float tot = wave_wmma_sum256(red, t);
      if (t == 0) results[c] = tot;
    }
    __syncthreads();
  }

  if (t == 0) {
    const float invB = 1.0f / (float)B_DIM;
    float loss = 0.0f;
#pragma unroll
    for (int f = 0; f < 5; ++f) {
      const float d = (results[f] - results[5 + f]) * invB;  // mean diff
      loss += d * d;
    }
    out[0] = loss;  // time-only features cancel exactly -> contribute 0
  }
}

extern "C" void kernel_launch(void* const* d_in, const int* in_sizes, int n_in,
                              void* d_out, int out_size, void* d_ws, size_t ws_size,
                              hipStream_t stream) {
  const float* real_p = (const float*)d_in[0];
  const float* gen_p  = (const float*)d_in[1];
  float* partials     = (float*)d_ws;  // 10 * 4096 floats = 160 KB

  sig_row_kernel<<<2 * B_DIM, NTHREADS, 0, stream>>>(real_p, gen_p, partials);
  sig_finalize_kernel<<<1, NTHREADS, 0, stream>>>(partials, (float*)d_out);
}